// GraphNet_49984829391383
// MI455X (gfx1250) — compile-verified
//
#include <hip/hip_runtime.h>

typedef __attribute__((ext_vector_type(2))) float v2f;
typedef __attribute__((ext_vector_type(8))) float v8f;

#define F_IN 128
#define H1   256
#define H2   64
#define NCLS 40
#define NPAD 48   // Wfc columns padded to multiple of 16

// ---------------- elementwise / degree kernels ----------------
__global__ void fill_kernel(float* __restrict__ p, float v, int n) {
  int i = blockIdx.x * blockDim.x + threadIdx.x;
  if (i < n) p[i] = v;
}

__global__ void deg_accum_kernel(const int* __restrict__ dst, float* __restrict__ deg, int E) {
  int e = blockIdx.x * blockDim.x + threadIdx.x;
  if (e < E) atomicAdd(&deg[dst[e]], 1.0f);
}

__global__ void rsqrt_kernel(float* __restrict__ p, int n) {
  int i = blockIdx.x * blockDim.x + threadIdx.x;
  if (i < n) p[i] = rsqrtf(p[i]);   // deg >= 1 always (self-loops)
}

template<int F>  // F must be a power of two (256 or 64)
__global__ void bias_init_kernel(float* __restrict__ out, const float* __restrict__ b, int total) {
  int i = blockIdx.x * blockDim.x + threadIdx.x;
  if (i < total) out[i] = b[i & (F - 1)];
}

__global__ void relu_kernel(float* __restrict__ p, int n) {
  int i = blockIdx.x * blockDim.x + threadIdx.x;
  if (i < n) p[i] = fmaxf(p[i], 0.0f);
}

// Zero-pad Wfc [H2 x NCLS] -> [H2 x NPAD]
__global__ void pad_wfc_kernel(const float* __restrict__ w, float* __restrict__ wp) {
  int i = blockIdx.x * blockDim.x + threadIdx.x;
  if (i < H2 * NPAD) {
    int r = i / NPAD, c = i % NPAD;
    wp[i] = (c < NCLS) ? w[r * NCLS + c] : 0.0f;
  }
}

// ---------------- edge-parallel normalized scatter-add ----------------
// out[dst] += dinv[src]*dinv[dst] * xw[src]  over E real edges + N self-loops.
// Feature index = thread within row -> coalesced xw reads and conflict-free
// per-feature f32 atomics. xw (102 MB / 26 MB) lives in the 192 MB L2.
template<int F, int LOGF>
__global__ void agg_kernel(const float* __restrict__ xw, const int* __restrict__ src,
                           const int* __restrict__ dst, const float* __restrict__ dinv,
                           float* __restrict__ out, int E, int N) {
  const int epb = 256 >> LOGF;                  // edges per block
  int tid = threadIdx.x;
  int f = tid & (F - 1);
  int eLocal = tid >> LOGF;
  long long e = (long long)blockIdx.x * epb + eLocal;
  long long total = (long long)E + (long long)N;
  if (e >= total) return;
  int s, d;
  if (e < E) { s = src[e]; d = dst[e]; }
  else       { s = d = (int)(e - E); }          // self-loop
  float norm = dinv[s] * dinv[d];
  atomicAdd(&out[(long long)d * F + f], norm * xw[(long long)s * F + f]);
}

// ---------------- fp32 WMMA GEMM: C[M,OUTW] = A[M,K] @ B[K,NN] (+bias) ----------------
// Compile-time K/NN/OUTW: no predicated loads in the hot loop.
// blockDim = 128 (4 waves). Wave w computes tile (mTile = blockIdx.y*4+w, nTile = blockIdx.x)
// with V_WMMA_F32_16X16X4_F32, K stepped by 4.
// A 16x4 operand (ISA 7.12.2): lanes 0-15 -> K={0,1}, lanes 16-31 -> K={2,3}, lane&15 = M.
// B 4x16 mirrors with lane&15 = N.  C/D: VGPR r = rows r / r+8 per lane half.
template<int K, int NN, int OUTW>
__global__ __launch_bounds__(128)
void gemm_wmma_f32(const float* __restrict__ A, const float* __restrict__ B,
                   const float* __restrict__ bias, float* __restrict__ C, int M) {
  int lane = threadIdx.x & 31;
  int wave = threadIdx.x >> 5;
  int mTile = blockIdx.y * 4 + wave;
  if (mTile * 16 >= M) return;                  // wave-uniform: EXEC all-ones at WMMA
  int nTile = blockIdx.x;

  int halfK = (lane >> 4) << 1;                 // 0 for lanes 0-15, 2 for lanes 16-31
  int row = mTile * 16 + (lane & 15);
  int col = nTile * 16 + (lane & 15);

  const float* __restrict__ arow = A + (long long)row * K + halfK;   // 8B aligned
  const float* __restrict__ bcol = B + (long long)halfK * NN + col;

  v8f acc = {0.f, 0.f, 0.f, 0.f, 0.f, 0.f, 0.f, 0.f};

#pragma unroll 8
  for (int k = 0; k < K; k += 4) {
    v2f a = *(const v2f*)(arow + k);            // K and K+1 (or K+2,K+3)
    v2f b;
    b.x = bcol[(long long)k * NN];
    b.y = bcol[(long long)k * NN + NN];
    // 8 args: (neg_a, A, neg_b, B, c_mod, C, reuse_a, reuse_b)
    acc = __builtin_amdgcn_wmma_f32_16x16x4_f32(false, a, false, b, (short)0, acc,
                                                false, false);
  }

  int rbase = mTile * 16 + ((lane >> 4) << 3);
  if (NN == OUTW || col < OUTW) {               // guard only in padded FC tile
    float badd = bias ? bias[col] : 0.0f;
#pragma unroll
    for (int r = 0; r < 8; ++r)
      C[(long long)(rbase + r) * OUTW + col] = acc[r] + badd;
  }
}

// ---------------- launcher ----------------
extern "C" void kernel_launch(void* const* d_in, const int* in_sizes, int n_in,
                              void* d_out, int out_size, void* d_ws, size_t ws_size,
                              hipStream_t stream) {
  const float* x   = (const float*)d_in[0];
  const int*   ei  = (const int*)d_in[1];
  const float* W1  = (const float*)d_in[2];
  const float* b1  = (const float*)d_in[3];
  const float* W2  = (const float*)d_in[4];
  const float* b2  = (const float*)d_in[5];
  const float* Wfc = (const float*)d_in[6];
  const float* bfc = (const float*)d_in[7];
  float* out = (float*)d_out;

  const int N = in_sizes[0] / F_IN;     // 100000
  const int E = in_sizes[1] / 2;        // 1600000
  const int* src = ei;
  const int* dst = ei + E;

  // Workspace layout (floats): dinv[N] | bufA[N*H1] | bufB[N*H1] | WfcPad[H2*NPAD]
  float* dinv   = (float*)d_ws;
  float* bufA   = dinv + N;                      // xw1, later xw2
  float* bufB   = bufA + (size_t)N * H1;         // h1,  later h2
  float* wfcPad = bufB + (size_t)N * H1;

  const long long totE = (long long)E + N;       // edges incl. self-loops

  // degrees -> dinv (in place)
  fill_kernel<<<(N + 255) / 256, 256, 0, stream>>>(dinv, 1.0f, N);   // self-loop
  deg_accum_kernel<<<(E + 255) / 256, 256, 0, stream>>>(dst, dinv, E);
  rsqrt_kernel<<<(N + 255) / 256, 256, 0, stream>>>(dinv, N);
  pad_wfc_kernel<<<(H2 * NPAD + 255) / 256, 256, 0, stream>>>(Wfc, wfcPad);

  // ---- layer 1: xw1 = x @ W1 ; h1 = b1 + scatter(norm*xw1) ; relu ----
  {
    dim3 grid(H1 / 16, (N / 16 + 3) / 4);
    gemm_wmma_f32<F_IN, H1, H1><<<grid, 128, 0, stream>>>(x, W1, nullptr, bufA, N);
  }
  bias_init_kernel<H1><<<N * (H1 / 256), 256, 0, stream>>>(bufB, b1, N * H1);
  agg_kernel<H1, 8><<<(unsigned)totE, 256, 0, stream>>>(bufA, src, dst, dinv, bufB, E, N);
  relu_kernel<<<N * (H1 / 256), 256, 0, stream>>>(bufB, N * H1);

  // ---- layer 2: xw2 = h1 @ W2 ; h2 = b2 + scatter(norm*xw2) ----
  {
    dim3 grid(H2 / 16, (N / 16 + 3) / 4);
    gemm_wmma_f32<H1, H2, H2><<<grid, 128, 0, stream>>>(bufB, W2, nullptr, bufA, N);
  }
  bias_init_kernel<H2><<<((size_t)N * H2 + 255) / 256, 256, 0, stream>>>(bufB, b2, N * H2);
  agg_kernel<H2, 6><<<(unsigned)((totE + 3) / 4), 256, 0, stream>>>(bufA, src, dst, dinv, bufB, E, N);

  // ---- FC: out = h2 @ WfcPad + bfc  (NN=48, real width 40 via store guard) ----
  {
    dim3 grid(NPAD / 16, (N / 16 + 3) / 4);
    gemm_wmma_f32<H2, NPAD, NCLS><<<grid, 128, 0, stream>>>(bufB, wfcPad, bfc, out, N);
  }
}